// KnowledgeAugmentedFusion_35931696399056
// MI455X (gfx1250) — compile-verified
//
#include <hip/hip_runtime.h>

// ---------------------------------------------------------------------------
// KnowledgeAugmentedFusion for MI455X (gfx1250).
// Dominant cost: s = einsum('bj,ijl,bl->bi') == GEMM [16 x 262144] x [262144 x 512]
// streaming 537MB of W3 exactly once -> HBM-bound (~23us @ 23.3 TB/s).
// V_WMMA_F32_16X16X4_F32, LDS double-buffered W3 tiles staged with
// global_load_async_to_lds_b128; A-side fully register-resident.
// ---------------------------------------------------------------------------

typedef float v2f   __attribute__((ext_vector_type(2)));
typedef float v8f   __attribute__((ext_vector_type(8)));
typedef float f32x4 __attribute__((ext_vector_type(4)));
typedef int   v4i   __attribute__((ext_vector_type(4)));

#define BATCH 16
#define DIM 512
#define KTOT (512 * 512)          // 262144 flattened (j,l)
#define SPLITK 64
#define KSLICE (KTOT / SPLITK)    // 4096 -> 8 consecutive j values per block
#define KTILE 256
#define NTILES (KSLICE / KTILE)   // 16
#define LDS_STRIDE (KTILE + 4)    // +4 words -> conflict-free striped reads

#if defined(__has_builtin)
#if __has_builtin(__builtin_amdgcn_global_load_async_to_lds_b128) && \
    __has_builtin(__builtin_amdgcn_s_wait_asynccnt)
#define HAVE_ASYNC_LDS 1
#endif
#endif

#if defined(HAVE_ASYNC_LDS)
// signature (from hipcc diagnostics):
//   void __builtin_amdgcn_global_load_async_to_lds_b128(v4i* src,
//        __attribute__((address_space(3))) v4i* dst, int offset, int cpol)
#define GSRC(p) ((v4i*)(p))
#define LDST(p) ((__attribute__((address_space(3))) v4i*)(p))
#endif

__global__ __launch_bounds__(256) void zero_f32(float* p, int n) {
    int i = blockIdx.x * 256 + threadIdx.x;
    if (i < n) p[i] = 0.0f;
}

// out[b, i] = dot(X[b, :], W[i, :]) + bias[i];  one block per output column i.
template <int IN>
__global__ __launch_bounds__(256) void proj_kernel(const float* __restrict__ X,
                                                   const float* __restrict__ W,
                                                   const float* __restrict__ bias,
                                                   float* __restrict__ out) {
    __shared__ float red[8 * BATCH];
    const int i = blockIdx.x;
    const int tid = threadIdx.x;
    const int lane = tid & 31;
    const int wave = tid >> 5;

    float part[BATCH];
#pragma unroll
    for (int b = 0; b < BATCH; ++b) part[b] = 0.0f;

    for (int c = tid; c < IN; c += 256) {
        float w = W[(long)i * IN + c];
#pragma unroll
        for (int b = 0; b < BATCH; ++b) part[b] += w * X[b * IN + c];
    }
#pragma unroll
    for (int b = 0; b < BATCH; ++b) {
        float v = part[b];
#pragma unroll
        for (int off = 16; off > 0; off >>= 1) v += __shfl_down(v, off, 32);
        if (lane == 0) red[wave * BATCH + b] = v;
    }
    __syncthreads();
    if (tid < BATCH) {
        float sum = 0.0f;
#pragma unroll
        for (int w = 0; w < 8; ++w) sum += red[w * BATCH + tid];
        out[tid * DIM + i] = sum + bias[i];
    }
}

// s[b, i0+n] += sum_{q in slice} (t[b,j]*k[b,l]) * W3[i0+n, q],  q = j*512 + l
__global__ __launch_bounds__(256) void trilinear_wmma(const float* __restrict__ W3,
                                                      const float* __restrict__ t,
                                                      const float* __restrict__ k,
                                                      float* __restrict__ s) {
    __shared__ float tile[2][16 * LDS_STRIDE];
    __shared__ float red[8 * 256];

    const int tid  = threadIdx.x;
    const int lane = tid & 31;
    const int wave = tid >> 5;
    const int it    = blockIdx.x & 31;  // i-tile: 16 rows of W3
    const int slice = blockIdx.x >> 5;  // split-K slice
    const int i0 = it * 16;
    const long kbeg = (long)slice * KSLICE;   // 512-aligned
    const int jbase = (int)(kbeg >> 9);       // 8 consecutive j per slice

    const int brow = lane & 15;  // A-matrix M (= batch b) per ISA layout
    const int half = lane >> 4;  // lanes 16..31 carry K = 2,3 of each window

    // ---- preload ALL A-side data into registers (t/k tiny, L2-resident) ----
    const float* krow = k + brow * DIM;
    const float* trow = t + brow * DIM + jbase;
    const int kcol = wave * 32 + half * 2;     // this wave's l-window offset
    v2f kva[8], kvb[8];
    float tv[8];
#pragma unroll
    for (int stp = 0; stp < 8; ++stp) {
        kva[stp] = *(const v2f*)(krow + kcol + stp * 4);         // l in [0,256)
        kvb[stp] = *(const v2f*)(krow + 256 + kcol + stp * 4);   // l in [256,512)
    }
#pragma unroll
    for (int jj = 0; jj < 8; ++jj) tv[jj] = trow[jj];

    // ---- W3 staging geometry: thread owns 4 float4s per tile --------------
    const int row0 = tid >> 6;            // rows row0, row0+4, row0+8, row0+12
    const int c4x4 = (tid & 63) * 4;      // 64 float4 per row, coalesced
    const float* g0 = W3 + (long)(i0 + row0) * KTOT + kbeg + c4x4;
    const int l0 = row0 * LDS_STRIDE + c4x4;

    v8f acc = {};
    f32x4 rg[4];
    (void)rg;

    // ---- prime tile 0 into buffer 0 ---------------------------------------
#if defined(HAVE_ASYNC_LDS)
#pragma unroll
    for (int rep = 0; rep < 4; ++rep)
        __builtin_amdgcn_global_load_async_to_lds_b128(
            GSRC(g0 + (long)rep * 4 * KTOT),
            LDST(&tile[0][l0 + rep * 4 * LDS_STRIDE]), 0, 0);
    __builtin_amdgcn_s_wait_asynccnt(0);
#else
#pragma unroll
    for (int rep = 0; rep < 4; ++rep)
        rg[rep] = *(const f32x4*)(g0 + (long)rep * 4 * KTOT);
#pragma unroll
    for (int rep = 0; rep < 4; ++rep)
        *(f32x4*)(&tile[0][l0 + rep * 4 * LDS_STRIDE]) = rg[rep];
#endif
    __syncthreads();

    // ---- main pipeline: 16 tiles, double-buffered, 1 barrier per tile -----
#pragma unroll
    for (int tI = 0; tI < NTILES; ++tI) {
        const int cur = tI & 1;

        // issue next tile's global loads (land in buf[1-cur])
#if defined(HAVE_ASYNC_LDS)
        if (tI + 1 < NTILES) {
#pragma unroll
            for (int rep = 0; rep < 4; ++rep)
                __builtin_amdgcn_global_load_async_to_lds_b128(
                    GSRC(g0 + (long)rep * 4 * KTOT + (tI + 1) * KTILE),
                    LDST(&tile[1 - cur][l0 + rep * 4 * LDS_STRIDE]), 0, 0);
        }
#else
        if (tI + 1 < NTILES) {
#pragma unroll
            for (int rep = 0; rep < 4; ++rep)
                rg[rep] = *(const f32x4*)(g0 + (long)rep * 4 * KTOT + (tI + 1) * KTILE);
        }
#endif
        // prefetch 2 tiles ahead to cover HBM latency (global_prefetch_b8)
        if (tI + 2 < NTILES) {
#pragma unroll
            for (int rep = 0; rep < 4; ++rep)
                __builtin_prefetch(g0 + (long)rep * 4 * KTOT + (tI + 2) * KTILE, 0, 1);
        }

        // compute tile tI: j = jbase + (tI>>1), l-window parity = tI&1
        const float tvc = tv[tI >> 1];
#pragma unroll
        for (int stp = 0; stp < 8; ++stp) {
            const v2f kv = (tI & 1) ? kvb[stp] : kva[stp];
            v2f a;
            a.x = tvc * kv.x;                  // A frag = t[b,j]*k[b,l..l+1]
            a.y = tvc * kv.y;
            const v2f bfrag = *(const v2f*)(
                &tile[cur][(lane & 15) * LDS_STRIDE + wave * 32 + stp * 4 + half * 2]);
            acc = __builtin_amdgcn_wmma_f32_16x16x4_f32(
                false, a, false, bfrag, (short)0, acc, false, false);
        }

#if defined(HAVE_ASYNC_LDS)
        __builtin_amdgcn_s_wait_asynccnt(0);
#else
        if (tI + 1 < NTILES) {
#pragma unroll
            for (int rep = 0; rep < 4; ++rep)
                *(f32x4*)(&tile[1 - cur][l0 + rep * 4 * LDS_STRIDE]) = rg[rep];
        }
#endif
        __syncthreads();
    }

    // ---- reduce the 8 per-wave accumulators, then split-K atomic into s ---
#pragma unroll
    for (int r = 0; r < 8; ++r) red[wave * 256 + r * 32 + lane] = acc[r];
    __syncthreads();
    {
        const int b  = tid >> 4;       // 0..15
        const int il = tid & 15;       // 0..15
        const int r  = b & 7;
        const int ln = il + ((b >> 3) << 4);
        float sum = 0.0f;
#pragma unroll
        for (int w = 0; w < 8; ++w) sum += red[w * 256 + r * 32 + ln];
        atomicAdd(s + b * DIM + i0 + il, sum);
    }
}

// f[b, i] = sum_d (v[b,d]*s[b,d]) * Wo[i,d] + bo[i]
__global__ __launch_bounds__(256) void outproj_kernel(const float* __restrict__ v,
                                                      const float* __restrict__ s,
                                                      const float* __restrict__ Wo,
                                                      const float* __restrict__ bo,
                                                      float* __restrict__ f) {
    __shared__ float red[8 * BATCH];
    const int i = blockIdx.x;
    const int tid = threadIdx.x;
    const int lane = tid & 31;
    const int wave = tid >> 5;

    float part[BATCH];
#pragma unroll
    for (int b = 0; b < BATCH; ++b) part[b] = 0.0f;

    for (int c = tid; c < DIM; c += 256) {
        float w = Wo[(long)i * DIM + c];
#pragma unroll
        for (int b = 0; b < BATCH; ++b)
            part[b] += w * v[b * DIM + c] * s[b * DIM + c];
    }
#pragma unroll
    for (int b = 0; b < BATCH; ++b) {
        float x = part[b];
#pragma unroll
        for (int off = 16; off > 0; off >>= 1) x += __shfl_down(x, off, 32);
        if (lane == 0) red[wave * BATCH + b] = x;
    }
    __syncthreads();
    if (tid < BATCH) {
        float sum = 0.0f;
#pragma unroll
        for (int w = 0; w < 8; ++w) sum += red[w * BATCH + tid];
        f[tid * DIM + i] = sum + bo[i];
    }
}

// LayerNorm over D=512, one block per batch row.
__global__ __launch_bounds__(256) void layernorm_kernel(const float* __restrict__ f,
                                                        const float* __restrict__ gamma,
                                                        const float* __restrict__ beta,
                                                        float* __restrict__ out) {
    __shared__ float r1[256];
    __shared__ float r2[256];
    const int b = blockIdx.x;
    const int tid = threadIdx.x;

    const float x0 = f[b * DIM + tid];
    const float x1 = f[b * DIM + tid + 256];
    r1[tid] = x0 + x1;
    r2[tid] = x0 * x0 + x1 * x1;
    __syncthreads();
    for (int off = 128; off > 0; off >>= 1) {
        if (tid < off) {
            r1[tid] += r1[tid + off];
            r2[tid] += r2[tid + off];
        }
        __syncthreads();
    }
    const float mu   = r1[0] * (1.0f / 512.0f);
    const float var  = r2[0] * (1.0f / 512.0f) - mu * mu;
    const float rinv = rsqrtf(var + 1e-5f);
    out[b * DIM + tid]       = (x0 - mu) * rinv * gamma[tid]       + beta[tid];
    out[b * DIM + tid + 256] = (x1 - mu) * rinv * gamma[tid + 256] + beta[tid + 256];
}

extern "C" void kernel_launch(void* const* d_in, const int* in_sizes, int n_in,
                              void* d_out, int out_size, void* d_ws, size_t ws_size,
                              hipStream_t stream) {
    const float* vis   = (const float*)d_in[0];   // [16, 2048]
    const float* txt   = (const float*)d_in[1];   // [16, 768]
    const float* kno   = (const float*)d_in[2];   // [16, 1024]
    const float* Wv    = (const float*)d_in[3];   // [512, 2048]
    const float* bv    = (const float*)d_in[4];
    const float* Wt    = (const float*)d_in[5];   // [512, 768]
    const float* bt    = (const float*)d_in[6];
    const float* Wk    = (const float*)d_in[7];   // [512, 1024]
    const float* bk    = (const float*)d_in[8];
    const float* W3    = (const float*)d_in[9];   // [512, 512, 512]
    const float* Wo    = (const float*)d_in[10];  // [512, 512]
    const float* bo    = (const float*)d_in[11];
    const float* gamma = (const float*)d_in[12];
    const float* beta  = (const float*)d_in[13];

    float* ws = (float*)d_ws;
    float* t_buf = ws;                 // [16, 512]
    float* k_buf = ws + 8192;          // [16, 512]
    float* v_buf = ws + 16384;         // [16, 512]
    float* s_buf = ws + 24576;         // [16, 512]
    float* f_buf = ws + 32768;         // [16, 512]

    zero_f32<<<(BATCH * DIM + 255) / 256, 256, 0, stream>>>(s_buf, BATCH * DIM);

    proj_kernel<2048><<<DIM, 256, 0, stream>>>(vis, Wv, bv, v_buf);
    proj_kernel<768><<<DIM, 256, 0, stream>>>(txt, Wt, bt, t_buf);
    proj_kernel<1024><<<DIM, 256, 0, stream>>>(kno, Wk, bk, k_buf);

    trilinear_wmma<<<32 * SPLITK, 256, 0, stream>>>(W3, t_buf, k_buf, s_buf);

    outproj_kernel<<<DIM, 256, 0, stream>>>(v_buf, s_buf, Wo, bo, f_buf);
    layernorm_kernel<<<BATCH, 256, 0, stream>>>(f_buf, gamma, beta, (float*)d_out);
}